// HyperGraphBasicConvolution_22866405884229
// MI455X (gfx1250) — compile-verified
//
#include <hip/hip_runtime.h>
#include <hip/hip_bf16.h>

// ---------------------------------------------------------------------------
// HyperGraph convolution for MI455X (gfx1250, wave32).
//   user_msg = spmm(U, user_emb)      [G=20000, D=128]
//   item_msg = spmm(I, item_emb)      [G, D]
//   msg      = [user_msg|item_msg] @ W_aggT + b        (fp32 WMMA 16x16x4)
//   norm_emb = spmm(F, msg)           [U+I=150000, D]
// d_out = norm_emb (150000*128) ++ msg (20000*128), fp32.
// ---------------------------------------------------------------------------

#define DIM      128
#define NGROUPS  20000
#define NUSERS   50000
#define NITEMS   100000
#define NUI      150000   // NUSERS + NITEMS

typedef __attribute__((ext_vector_type(2))) float v2f;
typedef __attribute__((ext_vector_type(8))) float v8f;

// ---------------------------------------------------------------------------
// Zero-fill (float4 stores); atomics accumulate so destinations must start 0.
// ---------------------------------------------------------------------------
__global__ void __launch_bounds__(256) zero_f4_kernel(float4* __restrict__ p, long n4) {
    long i = (long)blockIdx.x * blockDim.x + threadIdx.x;
    long stride = (long)gridDim.x * blockDim.x;
    float4 z; z.x = 0.f; z.y = 0.f; z.z = 0.f; z.w = 0.f;
    for (; i < n4; i += stride) p[i] = z;
}

// ---------------------------------------------------------------------------
// COO SpMM scatter: out[rows[e], :] += vals[e] * dense[cols[e], :]
// One wave32 per edge; each lane moves a float4 slice of the 128-wide row.
// fp32 atomics resolve in L2 (dense + accumulator working sets fit in 192MB L2).
// ---------------------------------------------------------------------------
__global__ void __launch_bounds__(256) spmm_scatter_kernel(
    const int* __restrict__ rows, const int* __restrict__ cols,
    const float* __restrict__ vals, const float* __restrict__ dense,
    float* __restrict__ out, int nnz) {
    long gid  = (long)blockIdx.x * blockDim.x + threadIdx.x;
    int  edge = (int)(gid >> 5);
    int  lane = (int)(gid & 31);
    if (edge >= nnz) return;
    int   r = rows[edge];
    int   c = cols[edge];
    float v = vals[edge];
    const float4 s = ((const float4*)(dense + (size_t)c * DIM))[lane];
    float* dst = out + (size_t)r * DIM + (size_t)lane * 4;
    unsafeAtomicAdd(dst + 0, v * s.x);
    unsafeAtomicAdd(dst + 1, v * s.y);
    unsafeAtomicAdd(dst + 2, v * s.z);
    unsafeAtomicAdd(dst + 3, v * s.w);
}

// ---------------------------------------------------------------------------
// Dense GEMM via V_WMMA_F32_16X16X4_F32 (exact fp32, K stepped by 4).
//   msg[g, n] = sum_k A[g, k] * W[n, k] + b[n],  A = [user_msg | item_msg]
// One wave computes one 16x16 (M x N) output tile.
// fp32 WMMA operand layouts (ISA 7.12.2):
//   A 16x4 : lane 0-15 -> M=lane, K={0,1}; lane 16-31 -> M=lane-16, K={2,3}
//   B 4x16 : lane 0-15 -> N=lane, K={0,1}; lane 16-31 -> N=lane-16, K={2,3}
//   C/D    : VGPR v, lanes 0-15 -> M=v, N=lane; lanes 16-31 -> M=v+8, N=lane-16
// Grid: blockIdx.x = M-tile (1250), wave-in-block (8) = N-tile (128/16).
// ---------------------------------------------------------------------------
__global__ void __launch_bounds__(256) gemm_wmma_kernel(
    const float* __restrict__ Au,   // [NGROUPS, 128] user_msg
    const float* __restrict__ Ai,   // [NGROUPS, 128] item_msg
    const float* __restrict__ W,    // [128, 256] row-major (W_agg)
    const float* __restrict__ bias, // [128]
    float* __restrict__ msg) {      // [NGROUPS, 128]
    const int lane   = threadIdx.x & 31;
    const int wave   = threadIdx.x >> 5;      // 0..7 -> N tile
    const int hi     = lane >> 4;             // 0 or 1
    const int l16    = lane & 15;
    const int m_base = blockIdx.x * 16;
    const int n_base = wave * 16;

    const int am    = m_base + l16;           // A row this lane supplies
    const int bn    = n_base + l16;           // B column this lane supplies
    const int koff  = hi * 2;                 // lanes 16-31 carry K+2, K+3

    const float* arow_u = Au + (size_t)am * DIM;
    const float* arow_i = Ai + (size_t)am * DIM;
    const float* wrow   = W  + (size_t)bn * (2 * DIM);

    v8f acc = {};
    // K = 0..127 : user half of the concat
    #pragma unroll
    for (int k = 0; k < DIM; k += 4) {
        v2f a, b;
        float2 af = *(const float2*)(arow_u + k + koff);
        float2 bf = *(const float2*)(wrow   + k + koff);
        a.x = af.x; a.y = af.y;
        b.x = bf.x; b.y = bf.y;
        acc = __builtin_amdgcn_wmma_f32_16x16x4_f32(
            false, a, false, b, (short)0, acc, false, false);
    }
    // K = 128..255 : item half of the concat
    #pragma unroll
    for (int k = 0; k < DIM; k += 4) {
        v2f a, b;
        float2 af = *(const float2*)(arow_i + k + koff);
        float2 bf = *(const float2*)(wrow + DIM + k + koff);
        a.x = af.x; a.y = af.y;
        b.x = bf.x; b.y = bf.y;
        acc = __builtin_amdgcn_wmma_f32_16x16x4_f32(
            false, a, false, b, (short)0, acc, false, false);
    }

    const float bv = bias[n_base + l16];
    #pragma unroll
    for (int v = 0; v < 8; ++v) {
        int m = m_base + v + hi * 8;
        msg[(size_t)m * DIM + n_base + l16] = acc[v] + bv;
    }
}

// ---------------------------------------------------------------------------
// Launch
// ---------------------------------------------------------------------------
extern "C" void kernel_launch(void* const* d_in, const int* in_sizes, int n_in,
                              void* d_out, int out_size, void* d_ws, size_t ws_size,
                              hipStream_t stream) {
    const float* user_emb = (const float*)d_in[0];
    const float* item_emb = (const float*)d_in[1];
    // d_in[2] = group_emb : unused by reference
    const int*   u_rows = (const int*)d_in[3];
    const int*   u_cols = (const int*)d_in[4];
    const float* u_vals = (const float*)d_in[5];
    const int*   i_rows = (const int*)d_in[6];
    const int*   i_cols = (const int*)d_in[7];
    const float* i_vals = (const float*)d_in[8];
    const int*   f_rows = (const int*)d_in[9];
    const int*   f_cols = (const int*)d_in[10];
    const float* f_vals = (const float*)d_in[11];
    const float* W_agg  = (const float*)d_in[12];
    const float* b_agg  = (const float*)d_in[13];

    const int nnz_u = in_sizes[3];
    const int nnz_i = in_sizes[6];
    const int nnz_f = in_sizes[9];

    float* ws_u     = (float*)d_ws;                    // [NGROUPS, DIM]
    float* ws_i     = ws_u + (size_t)NGROUPS * DIM;    // [NGROUPS, DIM]
    float* norm_out = (float*)d_out;                   // [NUI, DIM]
    float* msg_out  = norm_out + (size_t)NUI * DIM;    // [NGROUPS, DIM]

    // 1) zero accumulators (workspace partial msgs + norm_emb region)
    {
        long n4 = (long)2 * NGROUPS * DIM / 4;
        int blocks = (int)((n4 + 255) / 256);
        zero_f4_kernel<<<blocks, 256, 0, stream>>>((float4*)ws_u, n4);
    }
    {
        long n4 = (long)NUI * DIM / 4;
        int blocks = (int)((n4 + 255) / 256);
        zero_f4_kernel<<<blocks, 256, 0, stream>>>((float4*)norm_out, n4);
    }

    // 2) user/item hypergraph SpMMs into group accumulators
    {
        long threads = (long)nnz_u * 32;
        int blocks = (int)((threads + 255) / 256);
        spmm_scatter_kernel<<<blocks, 256, 0, stream>>>(
            u_rows, u_cols, u_vals, user_emb, ws_u, nnz_u);
    }
    {
        long threads = (long)nnz_i * 32;
        int blocks = (int)((threads + 255) / 256);
        spmm_scatter_kernel<<<blocks, 256, 0, stream>>>(
            i_rows, i_cols, i_vals, item_emb, ws_i, nnz_i);
    }

    // 3) msg = [user_msg | item_msg] @ W_aggT + b  (fp32 WMMA)
    gemm_wmma_kernel<<<NGROUPS / 16, 256, 0, stream>>>(
        ws_u, ws_i, W_agg, b_agg, msg_out);

    // 4) norm_emb = spmm(F, msg)
    {
        long threads = (long)nnz_f * 32;
        int blocks = (int)((threads + 255) / 256);
        spmm_scatter_kernel<<<blocks, 256, 0, stream>>>(
            f_rows, f_cols, f_vals, msg_out, norm_out, nnz_f);
    }
}